// Softmax_fi_30889404793374
// MI455X (gfx1250) — compile-verified
//
#include <hip/hip_runtime.h>
#include <stdint.h>

#define TPB 256
#define ROWLEN 2048u
#define NPAIR_ROWS 32768u                    // block b handles rows b and b+32768
#define HALF_ELEMS (NPAIR_ROWS * ROWLEN)     // 67108864 = N/2 (threefry counter pairing)
#define FLIP_THRESH 8389u                    // (bits>>9) < 8389  <=>  uniform(bits) < 0.001f
#define FLIP_MASK 0x40000000u                // bit 30

typedef __attribute__((ext_vector_type(4))) float v4f;   // native vector: NT-store legal

// JAX threefry2x32 (20 rounds), bit-exact key-injection schedule.
__host__ __device__ static inline void threefry2x32(
    uint32_t k0, uint32_t k1, uint32_t c0, uint32_t c1,
    uint32_t* o0, uint32_t* o1)
{
    const uint32_t ks2 = k0 ^ k1 ^ 0x1BD11BDAu;
    uint32_t x0 = c0 + k0;
    uint32_t x1 = c1 + k1;
#define TF_R(r) { x0 += x1; x1 = (x1 << (r)) | (x1 >> (32 - (r))); x1 ^= x0; }
    TF_R(13) TF_R(15) TF_R(26) TF_R(6)
    x0 += k1;  x1 += ks2 + 1u;
    TF_R(17) TF_R(29) TF_R(16) TF_R(24)
    x0 += ks2; x1 += k0 + 2u;
    TF_R(13) TF_R(15) TF_R(26) TF_R(6)
    x0 += k0;  x1 += k1 + 3u;
    TF_R(17) TF_R(29) TF_R(16) TF_R(24)
    x0 += k1;  x1 += ks2 + 4u;
    TF_R(13) TF_R(15) TF_R(26) TF_R(6)
    x0 += ks2; x1 += k0 + 5u;
#undef TF_R
    *o0 = x0; *o1 = x1;
}

__device__ static inline uint32_t flip_mask(uint32_t bits) {
    return ((bits >> 9) < FLIP_THRESH) ? FLIP_MASK : 0u;
}

__device__ static inline float flip_if(float v, uint32_t bits) {
    return __uint_as_float(__float_as_uint(v) ^ flip_mask(bits));
}

__global__ __launch_bounds__(TPB) void softmax_fi_kernel(
    const float* __restrict__ x, float* __restrict__ y,
    uint32_t k1a, uint32_t k1b, uint32_t k2a, uint32_t k2b,
    uint32_t k3a, uint32_t k3b)
{
    __shared__ __align__(16) float stage[2 * ROWLEN];   // 16 KB: both paired rows
    __shared__ float wsum[2][TPB / 32];
    __shared__ float part[2];

    const uint32_t r0  = blockIdx.x;                    // 0..32767
    const uint32_t tid = threadIdx.x;
    const uint32_t c   = tid * 8u;                      // column base, 8 floats/thread/row

    // global byte offsets (u32: tensor is 512 MB, second half offset +256 MB)
    const uint32_t g0 = (r0 * ROWLEN + c) * 4u;
    const uint32_t g1 = g0 + HALF_ELEMS * 4u;
    // raw LDS byte offsets: low 32 bits of the flat shared pointer ARE the LDS offset
    const uint32_t l0 = (uint32_t)(uintptr_t)&stage[c];
    const uint32_t l1 = (uint32_t)(uintptr_t)&stage[ROWLEN + c];
    const uint64_t xbase = (uint64_t)(uintptr_t)x;

    // CDNA5 async global->LDS staging (ASYNCcnt path), GVS addressing.
    // Input is streamed exactly once through the 192 MB L2 -> non-temporal hint.
    asm volatile(
        "global_load_async_to_lds_b128 %0, %1, %4 offset:0 th:TH_LOAD_NT\n\t"
        "global_load_async_to_lds_b128 %0, %1, %4 offset:16 th:TH_LOAD_NT\n\t"
        "global_load_async_to_lds_b128 %2, %3, %4 offset:0 th:TH_LOAD_NT\n\t"
        "global_load_async_to_lds_b128 %2, %3, %4 offset:16 th:TH_LOAD_NT"
        :
        : "v"(l0), "v"(g0), "v"(l1), "v"(g1), "s"(xbase)
        : "memory");
    asm volatile("s_wait_asynccnt 0x0" ::: "memory");

    const v4f a0 = *(const v4f*)&stage[c];
    const v4f a1 = *(const v4f*)&stage[c + 4u];
    const v4f b0 = *(const v4f*)&stage[ROWLEN + c];
    const v4f b1 = *(const v4f*)&stage[ROWLEN + c + 4u];

    const float va[8] = {a0.x, a0.y, a0.z, a0.w, a1.x, a1.y, a1.z, a1.w};
    const float vb[8] = {b0.x, b0.y, b0.z, b0.w, b1.x, b1.y, b1.z, b1.w};

    float e0[8], e1[8];
    float s0 = 0.f, s1 = 0.f;
    const uint32_t ibase = r0 * ROWLEN + c;             // flat index in first half

    // x_exp = exp(x); FI with k1. One threefry eval covers BOTH paired rows.
#pragma unroll
    for (int j = 0; j < 8; ++j) {
        float ex0 = __expf(va[j]);
        float ex1 = __expf(vb[j]);
        uint32_t o0, o1;
        threefry2x32(k1a, k1b, ibase + (uint32_t)j, ibase + (uint32_t)j + HALF_ELEMS, &o0, &o1);
        ex0 = flip_if(ex0, o0);
        ex1 = flip_if(ex1, o1);
        e0[j] = ex0; e1[j] = ex1;
        s0 += ex0;  s1 += ex1;
    }

    // Precompute y-FI (k3) flip masks BEFORE the barriers: this PRNG work is
    // independent of the partition, so early-arriving waves burn it here
    // instead of idling at s_barrier_wait.
    uint32_t m0[8], m1[8];
#pragma unroll
    for (int j = 0; j < 8; ++j) {
        uint32_t o0, o1;
        threefry2x32(k3a, k3b, ibase + (uint32_t)j, ibase + (uint32_t)j + HALF_ELEMS, &o0, &o1);
        m0[j] = flip_mask(o0);
        m1[j] = flip_mask(o1);
    }

    // wave32 shuffle reduction, then 8-wave LDS combine
#pragma unroll
    for (int off = 16; off > 0; off >>= 1) {
        s0 += __shfl_down(s0, off, 32);
        s1 += __shfl_down(s1, off, 32);
    }
    const uint32_t wave = tid >> 5, lane = tid & 31u;
    if (lane == 0) { wsum[0][wave] = s0; wsum[1][wave] = s1; }
    __syncthreads();
    if (tid == 0) {
        float t0 = 0.f, t1 = 0.f;
#pragma unroll
        for (int w = 0; w < TPB / 32; ++w) { t0 += wsum[0][w]; t1 += wsum[1][w]; }
        // partition FI with k2: counter pair (r0, r0+32768) serves both rows
        uint32_t o0, o1;
        threefry2x32(k2a, k2b, r0, r0 + NPAIR_ROWS, &o0, &o1);
        part[0] = flip_if(t0, o0);
        part[1] = flip_if(t1, o1);
    }
    __syncthreads();
    const float p0 = part[0];
    const float p1 = part[1];

    // y = x_exp / partition; apply precomputed k3 flips; streaming NT stores
    float r0out[8], r1out[8];
#pragma unroll
    for (int j = 0; j < 8; ++j) {
        float y0 = e0[j] / p0;
        float y1 = e1[j] / p1;
        r0out[j] = __uint_as_float(__float_as_uint(y0) ^ m0[j]);
        r1out[j] = __uint_as_float(__float_as_uint(y1) ^ m1[j]);
    }
    v4f* yo0 = (v4f*)(y + (size_t)r0 * ROWLEN + c);
    v4f* yo1 = (v4f*)(y + ((size_t)r0 + NPAIR_ROWS) * ROWLEN + c);
    const v4f w0 = {r0out[0], r0out[1], r0out[2], r0out[3]};
    const v4f w1 = {r0out[4], r0out[5], r0out[6], r0out[7]};
    const v4f w2 = {r1out[0], r1out[1], r1out[2], r1out[3]};
    const v4f w3 = {r1out[4], r1out[5], r1out[6], r1out[7]};
    __builtin_nontemporal_store(w0, yo0);
    __builtin_nontemporal_store(w1, yo0 + 1);
    __builtin_nontemporal_store(w2, yo1);
    __builtin_nontemporal_store(w3, yo1 + 1);
}

extern "C" void kernel_launch(void* const* d_in, const int* in_sizes, int n_in,
                              void* d_out, int out_size, void* d_ws, size_t ws_size,
                              hipStream_t stream)
{
    (void)in_sizes; (void)n_in; (void)d_ws; (void)ws_size; (void)out_size;
    const float* x = (const float*)d_in[0];
    float* y = (float*)d_out;

    // jax.random.key(42) -> (hi,lo) = (0,42); split(key,3): counts iota(6), half=3,
    // pairs (0,3),(1,4),(2,5); reshape(3,2) row-major -> k1,k2,k3.
    uint32_t o[6];
    threefry2x32(0u, 42u, 0u, 3u, &o[0], &o[3]);
    threefry2x32(0u, 42u, 1u, 4u, &o[1], &o[4]);
    threefry2x32(0u, 42u, 2u, 5u, &o[2], &o[5]);
    const uint32_t k1a = o[0], k1b = o[1];
    const uint32_t k2a = o[2], k2b = o[3];
    const uint32_t k3a = o[4], k3b = o[5];

    softmax_fi_kernel<<<dim3(NPAIR_ROWS), dim3(TPB), 0, stream>>>(
        x, y, k1a, k1b, k2a, k2b, k3a, k3b);
}